// SpatialEntropyGuidance_76184129896982
// MI455X (gfx1250) — compile-verified
//
#include <hip/hip_runtime.h>
#include <math.h>

typedef __attribute__((ext_vector_type(2))) float v2f;
typedef __attribute__((ext_vector_type(8))) float v8f;

#define B_   16
#define C_   256
#define HW_  4096

__device__ __forceinline__ float fsigmoid(float x) {
    return 1.0f / (1.0f + __expf(-x));
}

// ---------------- kernel 0: zero the 64 accumulator floats ----------------
__global__ void k_init(float* __restrict__ acc) {
    acc[threadIdx.x] = 0.0f;   // launched with 64 threads: norm2[32] ++ sums[32]
}

// ---------------- kernel 1: pooled[t][b][c] = mean over H*W ----------------
__global__ void k_pool(const float* __restrict__ vis, const float* __restrict__ text,
                       float* __restrict__ pooled) {
    const int bid = blockIdx.x;              // [0, 2*16*256)
    const int t   = bid >> 12;
    const int bc  = bid & 4095;
    const float* f = (t == 0 ? vis : text) + (size_t)bc * HW_;
    float s = 0.0f;
    for (int i = threadIdx.x; i < HW_; i += 256) s += f[i];
    #pragma unroll
    for (int off = 16; off > 0; off >>= 1) s += __shfl_xor(s, off, 32);
    __shared__ float red[8];
    if ((threadIdx.x & 31) == 0) red[threadIdx.x >> 5] = s;
    __syncthreads();
    if (threadIdx.x == 0) {
        float tot = 0.0f;
        #pragma unroll
        for (int i = 0; i < 8; ++i) tot += red[i];
        pooled[bid] = tot * (1.0f / (float)HW_);
    }
}

// ---------------- kernel 2: tiny MLP via V_WMMA_F32_16X16X4_F32 ----------------
// one wave per tensor; B=16 rows == one WMMA M-tile.
// A 16x4 f32 frag: lanes 0-15 M=0..15 {K0,K1}, lanes 16-31 M=0..15 {K2,K3}
// B 4x16 f32 frag: v0 = {K0 | K2}, v1 = {K1 | K3} across lane halves
// C/D 16x16 f32:  VGPR r: lanes0-15 -> (M=r, N=lane), lanes16-31 -> (M=r+8)
__global__ void k_mlp(const float* __restrict__ pooled,
                      const float* __restrict__ w1, const float* __restrict__ b1,
                      const float* __restrict__ w2, const float* __restrict__ b2,
                      float* __restrict__ cw) {
    const int t    = blockIdx.x;             // tensor 0/1
    const int lane = threadIdx.x;            // 0..31, wave32
    const int half = lane >> 4;
    const int l    = lane & 15;
    const float* P  = pooled + (size_t)t * B_ * C_;
    float*       CW = cw     + (size_t)t * B_ * C_;

    __shared__ float Hs[B_][64];

    // ---- layer 1: H = relu(P[16,256] @ w1[256,64] + b1) ----
    for (int nt = 0; nt < 4; ++nt) {
        const int n0 = nt * 16;
        v8f acc = {};
        for (int k = 0; k < C_; k += 4) {
            const int ka = k + half * 2;
            v2f a, b;
            a.x = P[l * C_ + ka + 0];
            a.y = P[l * C_ + ka + 1];
            b.x = w1[(ka + 0) * 64 + n0 + l];
            b.y = w1[(ka + 1) * 64 + n0 + l];
            acc = __builtin_amdgcn_wmma_f32_16x16x4_f32(false, a, false, b,
                                                        (short)0, acc, false, false);
        }
        #pragma unroll
        for (int r = 0; r < 8; ++r) {
            const int m = r + half * 8;
            const float v = acc[r] + b1[n0 + l];
            Hs[m][n0 + l] = v > 0.0f ? v : 0.0f;
        }
    }
    __syncthreads();

    // ---- layer 2: CW = sigmoid(H[16,64] @ w2[64,256] + b2) ----
    for (int nt = 0; nt < 16; ++nt) {
        const int n0 = nt * 16;
        v8f acc = {};
        for (int k = 0; k < 64; k += 4) {
            const int ka = k + half * 2;
            v2f a, b;
            a.x = Hs[l][ka + 0];
            a.y = Hs[l][ka + 1];
            b.x = w2[(ka + 0) * C_ + n0 + l];
            b.y = w2[(ka + 1) * C_ + n0 + l];
            acc = __builtin_amdgcn_wmma_f32_16x16x4_f32(false, a, false, b,
                                                        (short)0, acc, false, false);
        }
        #pragma unroll
        for (int r = 0; r < 8; ++r) {
            const int m = r + half * 8;
            CW[m * C_ + n0 + l] = fsigmoid(acc[r] + b2[n0 + l]);
        }
    }
}

// ---------------- kernel 3: se[b,hw] = sum_c ent(feat) * cw[b,c]; Σse, Σse² ----
__global__ void k_se(const float* __restrict__ vis, const float* __restrict__ text,
                     const float* __restrict__ cw, float* __restrict__ se,
                     float* __restrict__ norm2, float* __restrict__ sums) {
    const int bid   = blockIdx.x;            // [0, 2*16*16)
    const int t     = bid >> 8;
    const int b     = (bid >> 4) & 15;
    const int chunk = bid & 15;
    const int hw    = chunk * 256 + threadIdx.x;
    const float* f  = (t == 0 ? vis : text) + (size_t)b * C_ * HW_ + hw;
    const float* w  = cw + (size_t)(t * B_ + b) * C_;

    float acc = 0.0f;
    for (int c = 0; c < C_; ++c) {
        if (c + 4 < C_) __builtin_prefetch(f + (size_t)(c + 4) * HW_, 0, 0);
        const float x = f[(size_t)c * HW_];
        const float p = fsigmoid(x);
        const float e = __logf(p + 1e-9f) - (1.0f - p) * __logf(1.0f - p + 1e-9f);
        acc += e * w[c];
    }
    se[(size_t)(t * B_ + b) * HW_ + hw] = acc;

    float s1 = acc, s2 = acc * acc;
    #pragma unroll
    for (int off = 16; off > 0; off >>= 1) {
        s1 += __shfl_xor(s1, off, 32);
        s2 += __shfl_xor(s2, off, 32);
    }
    __shared__ float r1[8], r2[8];
    if ((threadIdx.x & 31) == 0) { r1[threadIdx.x >> 5] = s1; r2[threadIdx.x >> 5] = s2; }
    __syncthreads();
    if (threadIdx.x == 0) {
        float t1 = 0.0f, t2 = 0.0f;
        #pragma unroll
        for (int i = 0; i < 8; ++i) { t1 += r1[i]; t2 += r2[i]; }
        atomicAdd(&sums [t * B_ + b], t1);
        atomicAdd(&norm2[t * B_ + b], t2);
    }
}

// ---------------- kernel 4: normalized entropy outputs ----------------
__global__ void k_entout(const float* __restrict__ se, const float* __restrict__ norm2,
                         float* __restrict__ out_ent) {
    const int i  = blockIdx.x * 256 + threadIdx.x;   // [0, 2*16*4096)
    const int tb = i >> 12;
    const float n = fmaxf(sqrtf(norm2[tb]), 1e-12f);
    out_ent[i] = se[i] / n;
}

// ---------------- kernel 5: masked blend (float4 vectorized) ----------------
__global__ void k_fuse(const float4* __restrict__ vis4, const float4* __restrict__ text4,
                       const float* __restrict__ se, const float* __restrict__ sums,
                       float4* __restrict__ out4) {
    const size_t q  = (size_t)blockIdx.x * 256 + threadIdx.x;   // [0, 4194304)
    const size_t i  = q * 4;
    const int   b   = (int)(i >> 20);                 // / (C_*HW_)
    const int   hw  = (int)(i & (HW_ - 1));           // 4-aligned within row
    const float4 sv = *(const float4*)(se + (size_t)b * HW_ + hw);
    const float4 st = *(const float4*)(se + (size_t)(B_ + b) * HW_ + hw);
    const float thv = sums[b]      * (0.5f / (float)HW_);
    const float tht = sums[B_ + b] * (0.5f / (float)HW_);
    const float4 v  = vis4[q];
    const float4 x  = text4[q];
    float4 o;
    {
        const float vm = sv.x < thv ? 1.0f : 0.0f, tm = st.x < tht ? 1.0f : 0.0f;
        o.x = v.x * vm + x.x * tm * (1.0f - vm);
    }
    {
        const float vm = sv.y < thv ? 1.0f : 0.0f, tm = st.y < tht ? 1.0f : 0.0f;
        o.y = v.y * vm + x.y * tm * (1.0f - vm);
    }
    {
        const float vm = sv.z < thv ? 1.0f : 0.0f, tm = st.z < tht ? 1.0f : 0.0f;
        o.z = v.z * vm + x.z * tm * (1.0f - vm);
    }
    {
        const float vm = sv.w < thv ? 1.0f : 0.0f, tm = st.w < tht ? 1.0f : 0.0f;
        o.w = v.w * vm + x.w * tm * (1.0f - vm);
    }
    out4[q] = o;
}

extern "C" void kernel_launch(void* const* d_in, const int* in_sizes, int n_in,
                              void* d_out, int out_size, void* d_ws, size_t ws_size,
                              hipStream_t stream) {
    const float* vis  = (const float*)d_in[0];
    const float* text = (const float*)d_in[1];
    const float* w1   = (const float*)d_in[2];
    const float* b1   = (const float*)d_in[3];
    const float* w2   = (const float*)d_in[4];
    const float* b2   = (const float*)d_in[5];
    float* out = (float*)d_out;

    float* ws     = (float*)d_ws;
    float* pooled = ws;                 // 2*16*256   = 8192
    float* cw     = ws + 8192;          // 2*16*256   = 8192
    float* se     = ws + 16384;         // 2*16*4096  = 131072
    float* norm2  = ws + 147456;        // 32
    float* sums   = ws + 147488;        // 32  (contiguous after norm2)

    float* out_ent = out + (size_t)B_ * C_ * HW_;   // enhanced ++ vis_ent ++ text_ent

    k_init  <<<1,     64,  0, stream>>>(norm2);                         // zeros norm2+sums
    k_pool  <<<8192,  256, 0, stream>>>(vis, text, pooled);
    k_mlp   <<<2,     32,  0, stream>>>(pooled, w1, b1, w2, b2, cw);
    k_se    <<<512,   256, 0, stream>>>(vis, text, cw, se, norm2, sums);
    k_entout<<<512,   256, 0, stream>>>(se, norm2, out_ent);
    k_fuse  <<<16384, 256, 0, stream>>>((const float4*)vis, (const float4*)text,
                                        se, sums, (float4*)out);
}